// BlockModel_2267742732806
// MI455X (gfx1250) — compile-verified
//
#include <hip/hip_runtime.h>
#include <hip/hip_bf16.h>

#define N_NODES   100000
#define N_EDGES   1600000
#define N_SCORE   400000
#define NODE_F    32
#define EDGE_F    16
#define HID       128

typedef __attribute__((ext_vector_type(2))) float v2f;
typedef __attribute__((ext_vector_type(8))) float v8f;

// ---------------- utility kernels ----------------

__global__ void zero_f32(float* __restrict__ p, int n) {
  int i = blockIdx.x * blockDim.x + threadIdx.x;
  if (i < n) p[i] = 0.0f;
}

__global__ void zero_f32x4(float4* __restrict__ p, int n4) {
  int i = blockIdx.x * blockDim.x + threadIdx.x;
  if (i < n4) p[i] = make_float4(0.f, 0.f, 0.f, 0.f);
}

__global__ void degree_kernel(const int* __restrict__ src, const int* __restrict__ dst,
                              float* __restrict__ dout, float* __restrict__ din, int nE) {
  int i = blockIdx.x * blockDim.x + threadIdx.x;
  if (i < nE) {
    unsafeAtomicAdd(&dout[src[i]], 1.0f);   // global_atomic_add_f32, no return
    unsafeAtomicAdd(&din[dst[i]], 1.0f);
  }
}

__global__ void norm_kernel(float* __restrict__ p, int n) {
  int i = blockIdx.x * blockDim.x + threadIdx.x;
  if (i < n) {
    float d = p[i];
    d = d < 1.0f ? 1.0f : d;                // clip(deg, 1)
    p[i] = rsqrtf(d);                       // deg^-0.5
  }
}

// ---------------- WMMA f32 GEMM: C[M,HID] = f(pre .* A[M,K]) @ B[K,HID] ----------------
// One wave = one 16x16 tile. Block = 8 waves = one 16-row slab x all 128 cols.
// Epilogue: +bias, optional ReLU, optional per-row post-scale (out_norm).
template <int K, bool PRE, bool RELU, bool POST>
__global__ __launch_bounds__(256) void wmma_gemm(
    const float* __restrict__ A, const float* __restrict__ B,
    const float* __restrict__ bias, const float* __restrict__ pre,
    const float* __restrict__ post, float* __restrict__ C)
{
  const int wave = threadIdx.x >> 5;
  const int lane = threadIdx.x & 31;
  const int row0 = blockIdx.x << 4;       // 16 rows per block
  const int col0 = wave << 4;             // 16 cols per wave (8 waves -> 128)
  const int t  = lane & 15;               // A: row-in-tile / B,C: col-in-tile
  const int kb = (lane >> 4) << 1;        // K sub-offset {0,2} per lane half

  const float* Arow = A + (size_t)(row0 + t) * K;
  const float ps = PRE ? pre[row0 + t] : 1.0f;

  v8f acc = {};
#pragma unroll
  for (int k = 0; k < K; k += 4) {
    v2f a = *(const v2f*)(Arow + k + kb);            // A[M=t][k+kb], A[M=t][k+kb+1]
    if (PRE) { a.x *= ps; a.y *= ps; }
    v2f b;
    b.x = B[(size_t)(k + kb) * HID + col0 + t];      // B[k+kb][N=t]
    b.y = B[(size_t)(k + kb + 1) * HID + col0 + t];  // B[k+kb+1][N=t]
    acc = __builtin_amdgcn_wmma_f32_16x16x4_f32(
        /*neg_a=*/false, a, /*neg_b=*/false, b,
        /*c_mod=*/(short)0, acc, /*reuse_a=*/false, /*reuse_b=*/false);
  }

  const float bv = bias[col0 + t];
  const int mb = (lane >> 4) << 3;                   // row half: 0 or 8
#pragma unroll
  for (int r = 0; r < 8; ++r) {
    const int row = row0 + mb + r;
    float v = acc[r] + bv;
    if (RELU) v = v > 0.0f ? v : 0.0f;
    if (POST) v *= post[row];
    C[(size_t)row * HID + col0 + t] = v;
  }
}

// ---------------- edge scatter-add (message passing) ----------------
// h is already pre-scaled by out_norm. 32 lanes per edge, float4 per lane.
__global__ __launch_bounds__(256) void scatter_kernel(
    const float* __restrict__ h, float* __restrict__ agg,
    const int* __restrict__ src, const int* __restrict__ dst, int nE)
{
  size_t tid = (size_t)blockIdx.x * blockDim.x + threadIdx.x;
  int e = (int)(tid >> 5);
  int f = (int)(tid & 31) << 2;
  if (e < nE) {
    const int s = src[e], d = dst[e];
    const float4 v = *(const float4*)(h + (size_t)s * HID + f);
    float* p = agg + (size_t)d * HID + f;
    unsafeAtomicAdd(p + 0, v.x);
    unsafeAtomicAdd(p + 1, v.y);
    unsafeAtomicAdd(p + 2, v.z);
    unsafeAtomicAdd(p + 3, v.w);
  }
}

// ---------------- scorer precompute: wep = We @ Wp3 (16), c0 = bp + be . Wp3 ----------------
__global__ void scorer_pre(const float* __restrict__ We, const float* __restrict__ be,
                           const float* __restrict__ Wp, const float* __restrict__ bp,
                           float* __restrict__ wep, float* __restrict__ c0)
{
  int t = threadIdx.x;
  if (t < EDGE_F) {
    float s = 0.0f;
    for (int k = 0; k < HID; ++k) s += We[t * HID + k] * Wp[2 * HID + k];
    wep[t] = s;
  } else if (t == EDGE_F) {
    float s = bp[0];
    for (int k = 0; k < HID; ++k) s += be[k] * Wp[2 * HID + k];
    c0[0] = s;
  }
}

// ---------------- edge scorer: out = h[src].Wp1 + h[dst].Wp2 + e.wep + c0 ----------------
__global__ __launch_bounds__(256) void scorer_kernel(
    const float* __restrict__ h, const float* __restrict__ e,
    const float* __restrict__ Wp, const float* __restrict__ wep,
    const float* __restrict__ c0,
    const int* __restrict__ ssrc, const int* __restrict__ sdst,
    float* __restrict__ out, int nE)
{
  size_t tid = (size_t)blockIdx.x * blockDim.x + threadIdx.x;
  int i = (int)(tid >> 5);
  int lane = (int)(tid & 31);
  int f = lane << 2;
  if (i < nE) {
    const int s = ssrc[i], d = sdst[i];
    const float4 hs = *(const float4*)(h + (size_t)s * HID + f);
    const float4 hd = *(const float4*)(h + (size_t)d * HID + f);
    const float4 w1 = *(const float4*)(Wp + f);
    const float4 w2 = *(const float4*)(Wp + HID + f);
    float p = hs.x * w1.x + hs.y * w1.y + hs.z * w1.z + hs.w * w1.w
            + hd.x * w2.x + hd.y * w2.y + hd.z * w2.z + hd.w * w2.w;
    if (lane < 4) {
      const float4 ev = *(const float4*)(e + (size_t)i * EDGE_F + f);
      const float4 w3 = *(const float4*)(wep + f);
      p += ev.x * w3.x + ev.y * w3.y + ev.z * w3.z + ev.w * w3.w;
    }
#pragma unroll
    for (int off = 16; off > 0; off >>= 1) p += __shfl_xor(p, off, 32);
    if (lane == 0) out[i] = p + c0[0];
  }
}

// ---------------- host launch ----------------

extern "C" void kernel_launch(void* const* d_in, const int* in_sizes, int n_in,
                              void* d_out, int out_size, void* d_ws, size_t ws_size,
                              hipStream_t stream) {
  const float* x   = (const float*)d_in[0];
  const float* e   = (const float*)d_in[1];
  const float* Wn  = (const float*)d_in[2];
  const float* bn  = (const float*)d_in[3];
  const float* We  = (const float*)d_in[4];
  const float* be  = (const float*)d_in[5];
  const float* Wg  = (const float*)d_in[6];   // [3,128,128]
  const float* bg  = (const float*)d_in[7];   // [3,128]
  const float* Wp  = (const float*)d_in[8];   // [384,1]
  const float* bp  = (const float*)d_in[9];
  const int* edge_src  = (const int*)d_in[10];
  const int* edge_dst  = (const int*)d_in[11];
  const int* score_src = (const int*)d_in[12];
  const int* score_dst = (const int*)d_in[13];
  float* out = (float*)d_out;

  // workspace layout (floats)
  float* ws       = (float*)d_ws;
  float* out_norm = ws;                                   // N_NODES
  float* in_norm  = ws + N_NODES;                         // N_NODES
  float* hA       = ws + 2 * (size_t)N_NODES;             // N_NODES*HID
  float* hB       = hA + (size_t)N_NODES * HID;           // N_NODES*HID
  float* wep      = hB + (size_t)N_NODES * HID;           // 16
  float* c0       = wep + 16;                             // 1

  const int T = 256;
  const size_t NH = (size_t)N_NODES * HID;                // 12.8M floats

  // 1) degrees -> norms (both arrays are contiguous: one pass each)
  zero_f32<<<(2 * N_NODES + T - 1) / T, T, 0, stream>>>(out_norm, 2 * N_NODES);
  degree_kernel<<<(N_EDGES + T - 1) / T, T, 0, stream>>>(edge_src, edge_dst,
                                                         out_norm, in_norm, N_EDGES);
  norm_kernel<<<(2 * N_NODES + T - 1) / T, T, 0, stream>>>(out_norm, 2 * N_NODES);

  // 2) node encoder: hA = (x @ Wn + bn) * out_norm   (pre-scaled message source)
  wmma_gemm<NODE_F, false, false, true>
      <<<N_NODES / 16, T, 0, stream>>>(x, Wn, bn, nullptr, out_norm, hA);

  // 3) scorer precompute (independent; overlaps nothing it shouldn't)
  scorer_pre<<<1, 32, 0, stream>>>(We, be, Wp, bp, wep, c0);

  // 4) three GraphConv layers
  for (int l = 0; l < 3; ++l) {
    zero_f32x4<<<(int)((NH / 4 + T - 1) / T), T, 0, stream>>>((float4*)hB, (int)(NH / 4));
    scatter_kernel<<<(int)(((size_t)N_EDGES * 32 + T - 1) / T), T, 0, stream>>>(
        hA, hB, edge_src, edge_dst, N_EDGES);
    const float* W = Wg + (size_t)l * HID * HID;
    const float* b = bg + (size_t)l * HID;
    if (l < 2) {  // output feeds another conv: fold out_norm into epilogue
      wmma_gemm<HID, true, true, true>
          <<<N_NODES / 16, T, 0, stream>>>(hB, W, b, in_norm, out_norm, hA);
    } else {      // final layer: plain relu output for the scorer
      wmma_gemm<HID, true, true, false>
          <<<N_NODES / 16, T, 0, stream>>>(hB, W, b, in_norm, nullptr, hA);
    }
  }

  // 5) edge scorer
  scorer_kernel<<<(int)(((size_t)N_SCORE * 32 + T - 1) / T), T, 0, stream>>>(
      hA, e, Wp, wep, c0, score_src, score_dst, out, N_SCORE);
}